// GraphSAGE_71751723647376
// MI455X (gfx1250) — compile-verified
//
#include <hip/hip_runtime.h>
#include <hip/hip_bf16.h>
#include <math.h>

typedef __attribute__((ext_vector_type(2))) float v2f;
typedef __attribute__((ext_vector_type(8))) float v8f;

#define N_NODES 50000
#define N_EDGES 800000
#define IN_CH 128
#define HID 64
#define OUT_CH 40
#define L2_STRIDE 48   // padded logits stride (3 col tiles of 16)

// ---------------- utility: zero a float range (grid-stride) ----------------
__global__ void zero_f32(float* __restrict__ p, int n) {
    int stride = gridDim.x * blockDim.x;
    for (int i = blockIdx.x * blockDim.x + threadIdx.x; i < n; i += stride)
        p[i] = 0.0f;
}

// ---------------- edge aggregation, 128 channels: 1 wave per edge ----------
__global__ void agg_128(const float* __restrict__ x,
                        const long long* __restrict__ src,
                        const long long* __restrict__ dst,
                        float* __restrict__ sum, float* __restrict__ cnt) {
    int t = blockIdx.x * blockDim.x + threadIdx.x;
    int e = t >> 5;
    if (e >= N_EDGES) return;
    int lane = t & 31;
    int s = (int)src[e];
    int d = (int)dst[e];
    const float4 v = ((const float4*)(x + (size_t)s * IN_CH))[lane];
    float* sb = sum + (size_t)d * IN_CH + lane * 4;
    atomicAdd(sb + 0, v.x);
    atomicAdd(sb + 1, v.y);
    atomicAdd(sb + 2, v.z);
    atomicAdd(sb + 3, v.w);
    if (lane == 0) atomicAdd(cnt + d, 1.0f);
}

// ---------------- edge aggregation, 64 channels: 1 wave per edge -----------
__global__ void agg_64(const float* __restrict__ h,
                       const long long* __restrict__ src,
                       const long long* __restrict__ dst,
                       float* __restrict__ sum) {
    int t = blockIdx.x * blockDim.x + threadIdx.x;
    int e = t >> 5;
    if (e >= N_EDGES) return;
    int lane = t & 31;
    int s = (int)src[e];
    int d = (int)dst[e];
    const float2 v = ((const float2*)(h + (size_t)s * HID))[lane];
    float* sb = sum + (size_t)d * HID + lane * 2;
    atomicAdd(sb + 0, v.x);
    atomicAdd(sb + 1, v.y);
}

// ---------------- fused SAGE GEMM via V_WMMA_F32_16X16X4_F32 ---------------
// out[m, n] = relu?( (sum[m,:]/max(cnt[m],1)) @ Wl[n,:] + xin[m,:] @ Wr[n,:] + bias[n] )
// One wave computes a 16x16 output tile. Wave-uniform tile guard keeps
// EXEC == all-ones around the WMMA ops (ISA requirement).
__global__ void sage_wmma_gemm(const float* __restrict__ sum,
                               const float* __restrict__ cnt,
                               const float* __restrict__ xin,
                               const float* __restrict__ Wl,
                               const float* __restrict__ Wr,
                               const float* __restrict__ bias,
                               float* __restrict__ out,
                               int K, int nOut, int outStride,
                               int nColTiles, int totalWaves, int doRelu) {
    int wave = blockIdx.x * (blockDim.x >> 5) + (threadIdx.x >> 5);
    if (wave >= totalWaves) return;              // wave-uniform

    int tileRow = wave / nColTiles;
    int tileCol = wave % nColTiles;
    int lane = threadIdx.x & 31;
    int m    = lane & 15;
    int kh   = (lane >> 4) << 1;                 // lanes 0-15 -> K=k,k+1; 16-31 -> K=k+2,k+3

    int row = tileRow * 16 + m;                  // A-fragment row (both halves)
    int col = tileCol * 16 + m;                  // B-fragment column
    bool colOk = col < nOut;
    int colC = colOk ? col : 0;                  // clamp to avoid OOB reads

    float inv = 1.0f / fmaxf(cnt[row], 1.0f);
    const float* sRow  = sum + (size_t)row  * K;
    const float* xRow  = xin + (size_t)row  * K;
    const float* wlRow = Wl  + (size_t)colC * K;
    const float* wrRow = Wr  + (size_t)colC * K;

    v8f c = {0.f, 0.f, 0.f, 0.f, 0.f, 0.f, 0.f, 0.f};

    for (int k = 0; k < K; k += 4) {
        int kk = k + kh;
        v2f am, ax, bl, br;
        am[0] = sRow[kk]     * inv;
        am[1] = sRow[kk + 1] * inv;
        ax[0] = xRow[kk];
        ax[1] = xRow[kk + 1];
        float bl0 = wlRow[kk], bl1 = wlRow[kk + 1];
        float br0 = wrRow[kk], br1 = wrRow[kk + 1];
        bl[0] = colOk ? bl0 : 0.0f;
        bl[1] = colOk ? bl1 : 0.0f;
        br[0] = colOk ? br0 : 0.0f;
        br[1] = colOk ? br1 : 0.0f;
        // D = A*B + C, f32 exact
        c = __builtin_amdgcn_wmma_f32_16x16x4_f32(false, am, false, bl,
                                                  (short)0, c, false, false);
        c = __builtin_amdgcn_wmma_f32_16x16x4_f32(false, ax, false, br,
                                                  (short)0, c, false, false);
    }

    // C/D layout: VGPR r -> lanes 0-15: (M=r, N=lane); lanes 16-31: (M=r+8, N=lane-16)
    int n = tileCol * 16 + m;
    int mBase = tileRow * 16 + ((lane >> 4) << 3);
    if (n < nOut) {
        float b = bias[n];
        #pragma unroll
        for (int r = 0; r < 8; ++r) {
            float v = c[r] + b;
            if (doRelu) v = fmaxf(v, 0.0f);
            out[(size_t)(mBase + r) * outStride + n] = v;
        }
    }
}

// ---------------- log_softmax over 40 classes, thread per row --------------
__global__ void log_softmax40(const float* __restrict__ logits,
                              float* __restrict__ out) {
    int r = blockIdx.x * blockDim.x + threadIdx.x;
    if (r >= N_NODES) return;
    const float* p = logits + (size_t)r * L2_STRIDE;
    float mx = p[0];
    #pragma unroll
    for (int c = 1; c < OUT_CH; ++c) mx = fmaxf(mx, p[c]);
    float s = 0.0f;
    #pragma unroll
    for (int c = 0; c < OUT_CH; ++c) s += __expf(p[c] - mx);
    float l = __logf(s);
    float* o = out + (size_t)r * OUT_CH;
    #pragma unroll
    for (int c = 0; c < OUT_CH; ++c) o[c] = p[c] - mx - l;
}

extern "C" void kernel_launch(void* const* d_in, const int* in_sizes, int n_in,
                              void* d_out, int out_size, void* d_ws, size_t ws_size,
                              hipStream_t stream) {
    const float*     x   = (const float*)d_in[0];
    const long long* ei  = (const long long*)d_in[1];   // int64 [2, E]
    const float*     W1l = (const float*)d_in[2];
    const float*     b1  = (const float*)d_in[3];
    const float*     W1r = (const float*)d_in[4];
    const float*     W2l = (const float*)d_in[5];
    const float*     b2  = (const float*)d_in[6];
    const float*     W2r = (const float*)d_in[7];
    float* out = (float*)d_out;

    const long long* src = ei;
    const long long* dst = ei + N_EDGES;

    // Workspace layout (floats):
    //   [0 .. 6.4M)      sum1  (50000*128)   -- later reused:
    //                      [0 .. 3.2M)   sum2 (50000*64)
    //                      [3.2M .. 5.6M) logits (50000*48)
    //   [6.4M .. 6.45M)  cnt   (50000)
    //   [6.45M .. 9.65M) h     (50000*64)
    float* f      = (float*)d_ws;
    float* sum1   = f;
    float* cnt    = f + (size_t)N_NODES * IN_CH;
    float* h      = cnt + N_NODES;
    float* sum2   = f;
    float* logits = f + (size_t)N_NODES * HID;

    // 1) zero sum1 + cnt (contiguous)
    zero_f32<<<4096, 256, 0, stream>>>(sum1, N_NODES * IN_CH + N_NODES);

    // 2) layer-1 aggregation (wave per edge, 128 ch)
    agg_128<<<(N_EDGES * 32) / 256, 256, 0, stream>>>(x, src, dst, sum1, cnt);

    // 3) layer-1 fused WMMA GEMM + bias + relu -> h
    {
        int nct = HID / 16;                       // 4 col tiles
        int tw  = (N_NODES / 16) * nct;           // 12500 waves
        sage_wmma_gemm<<<(tw + 3) / 4, 128, 0, stream>>>(
            sum1, cnt, x, W1l, W1r, b1, h, IN_CH, HID, HID, nct, tw, 1);
    }

    // 4) zero sum2 (aliases dead sum1 region; cnt untouched)
    zero_f32<<<4096, 256, 0, stream>>>(sum2, N_NODES * HID);

    // 5) layer-2 aggregation (wave per edge, 64 ch)
    agg_64<<<(N_EDGES * 32) / 256, 256, 0, stream>>>(h, src, dst, sum2);

    // 6) layer-2 fused WMMA GEMM + bias -> logits (stride 48, cols >= 40 unused)
    {
        int nct = 3;                              // ceil(40/16)
        int tw  = (N_NODES / 16) * nct;           // 9375 waves
        sage_wmma_gemm<<<(tw + 3) / 4, 128, 0, stream>>>(
            sum2, cnt, h, W2l, W2r, b2, logits, HID, OUT_CH, L2_STRIDE, nct, tw, 0);
    }

    // 7) log_softmax -> d_out [50000, 40]
    log_softmax40<<<(N_NODES + 255) / 256, 256, 0, stream>>>(logits, out);
}